// LabelSmoothingLoss_46755013984641
// MI455X (gfx1250) — compile-verified
//
#include <hip/hip_runtime.h>
#include <stdint.h>

// Label-smoothing loss, B=4096 x C=50257 f32 logits.
// Bandwidth-bound: 823 MB read once => ~35us floor @ 23.3 TB/s.
// CDNA5 async global->LDS copies (3-deep pipelined, non-temporal) feed a
// two-pass-per-tile softmax so each element costs 1 HBM read + 1 v_exp_f32.

#define THREADS 256
#define TILE    4096           // floats per tile (16 KB)
#define CHUNKS  4              // float4 chunks per thread per tile
#define PIPE    3              // tiles in flight

__device__ __forceinline__ float fast_exp2(float x) {
#if __has_builtin(__builtin_amdgcn_exp2f)
  return __builtin_amdgcn_exp2f(x);
#else
  return exp2f(x);
#endif
}
__device__ __forceinline__ float fast_log2(float x) {
#if __has_builtin(__builtin_amdgcn_logf)
  return __builtin_amdgcn_logf(x);
#else
  return log2f(x);
#endif
}

// CDNA5 async copy: GLOBAL_LOAD_ASYNC_TO_LDS_B128, GVS mode (SGPR64 base +
// VGPR32 byte offset; VDST VGPR holds LDS byte address). Non-temporal: the
// stream has zero reuse, keep it out of L2's way.
__device__ __forceinline__ void async_load_b128(unsigned lds_addr, unsigned goff,
                                                unsigned long long base) {
  asm volatile("global_load_async_to_lds_b128 %0, %1, %2 th:TH_LOAD_NT"
               :
               : "v"(lds_addr), "v"(goff), "s"(base)
               : "memory");
}

#define WAIT_ASYNC_8() asm volatile("s_wait_asynccnt 8" ::: "memory")
#define WAIT_ASYNC_4() asm volatile("s_wait_asynccnt 4" ::: "memory")
#define WAIT_ASYNC_0() asm volatile("s_wait_asynccnt 0" ::: "memory")
#define WAIT_DS_0()    asm volatile("s_wait_dscnt 0"    ::: "memory")

// C_STATIC > 0: compile-time column count (constant-folds tile count, wait
// ladder, tail trip count). C_STATIC == 0: runtime fallback.
template <int C_STATIC>
__global__ __launch_bounds__(THREADS)
void row_lse_kernel(const float* __restrict__ pred, const int* __restrict__ target,
                    float* __restrict__ row_loss, int C_rt) {
  __shared__ __align__(16) float tiles[PIPE * TILE];
  __shared__ float red_m[THREADS / 32];
  __shared__ float red_s[THREADS / 32];

  const int C = (C_STATIC > 0) ? C_STATIC : C_rt;

  const int   row  = blockIdx.x;
  const int   tid  = threadIdx.x;
  const float* rowp = pred + (size_t)row * (size_t)C;
  const unsigned long long base = (unsigned long long)rowp;
  const unsigned lds_base = (unsigned)(unsigned long long)(uintptr_t)&tiles[0];

  const int NT = C / TILE;        // full tiles (12 for C=50257)
  const int TS = NT * TILE;       // tail start

  auto issue = [&](int t) {
    const unsigned buf      = (unsigned)(t % PIPE) * (TILE * 4u);
    const unsigned tileByte = (unsigned)t * (TILE * 4u);
#pragma unroll
    for (int c = 0; c < CHUNKS; ++c) {
      const unsigned slot = (unsigned)(c * THREADS + tid) * 16u;  // float4 slot, bytes
      async_load_b128(lds_base + buf + slot, tileByte + slot, base);
    }
  };

  const float L2E = 1.4426950408889634f;   // log2(e)
  const float LN2 = 0.6931471805599453f;
  float m = -3.0e38f, s = 0.0f;            // running (max, sum exp(x-m))

  if (NT > 0) issue(0);
  if (NT > 1) issue(1);
  if (NT > 2) issue(2);

  for (int t = 0; t < NT; ++t) {
    const int rem = NT - 1 - t;            // tiles still pending after this one
    if (rem >= 2)      { WAIT_ASYNC_8(); } // t+1, t+2 may be in flight
    else if (rem == 1) { WAIT_ASYNC_4(); }
    else               { WAIT_ASYNC_0(); }

    const float4* lb = (const float4*)&tiles[(t % PIPE) * TILE];
    float v[4 * CHUNKS];
#pragma unroll
    for (int c = 0; c < CHUNKS; ++c) {
      float4 q = lb[c * THREADS + tid];    // read back own region: no barrier needed
      v[4 * c + 0] = q.x; v[4 * c + 1] = q.y;
      v[4 * c + 2] = q.z; v[4 * c + 3] = q.w;
    }

    float tmax = v[0];
#pragma unroll
    for (int i = 1; i < 4 * CHUNKS; ++i) tmax = fmaxf(tmax, v[i]);
    float tsum = 0.0f;
#pragma unroll
    for (int i = 0; i < 4 * CHUNKS; ++i) tsum += fast_exp2((v[i] - tmax) * L2E);

    const float nm = fmaxf(m, tmax);
    s = s    * fast_exp2((m    - nm) * L2E)
      + tsum * fast_exp2((tmax - nm) * L2E);
    m = nm;

    WAIT_DS_0();                           // LDS reads retired before buffer reuse
    if (t + PIPE < NT) issue(t + PIPE);
  }

  // Tail (C % TILE elements), online update straight from global.
  for (int j = TS + tid; j < C; j += THREADS) {
    const float x  = rowp[j];
    const float nm = fmaxf(m, x);
    s = s * fast_exp2((m - nm) * L2E) + fast_exp2((x - nm) * L2E);
    m = nm;
  }

  // Wave32 reduction of (m, s).
#pragma unroll
  for (int off = 16; off > 0; off >>= 1) {
    const float om = __shfl_xor(m, off, 32);
    const float os = __shfl_xor(s, off, 32);
    const float nm = fmaxf(m, om);
    s = s  * fast_exp2((m  - nm) * L2E)
      + os * fast_exp2((om - nm) * L2E);
    m = nm;
  }
  if ((tid & 31) == 0) { red_m[tid >> 5] = m; red_s[tid >> 5] = s; }
  __syncthreads();

  if (tid == 0) {
    float gm = red_m[0], gs = red_s[0];
#pragma unroll
    for (int w = 1; w < THREADS / 32; ++w) {
      const float om = red_m[w], os = red_s[w];
      const float nm = fmaxf(gm, om);
      gs = gs * fast_exp2((gm - nm) * L2E)
         + os * fast_exp2((om - nm) * L2E);
      gm = nm;
    }
    const float logZ = gm + LN2 * fast_log2(gs);

    const int t  = target[row];
    const int tl = (t - 1 < 0) ? 0 : t - 1;
    const int tr = (t + 1 > C - 1) ? C - 1 : t + 1;
    const float wl = (t == 0) ? 0.0f : ((t == C - 1) ? 0.2f : 0.1f);
    const float wr = (t == 0) ? 0.2f : ((t == C - 1) ? 0.0f : 0.1f);
    const float dot = 0.8f * rowp[t] + wl * rowp[tl] + wr * rowp[tr];
    row_loss[row] = logZ - dot;            // smoothing weights sum to 1
  }
}

__global__ __launch_bounds__(256)
void mean_kernel(const float* __restrict__ rl, float* __restrict__ out, int B) {
  __shared__ float red[8];
  const int tid = threadIdx.x;
  float acc = 0.0f;
  for (int i = tid; i < B; i += 256) acc += rl[i];
#pragma unroll
  for (int off = 16; off > 0; off >>= 1) acc += __shfl_xor(acc, off, 32);
  if ((tid & 31) == 0) red[tid >> 5] = acc;
  __syncthreads();
  if (tid == 0) {
    float t = 0.0f;
#pragma unroll
    for (int w = 0; w < 8; ++w) t += red[w];
    out[0] = t / (float)B;
  }
}

extern "C" void kernel_launch(void* const* d_in, const int* in_sizes, int n_in,
                              void* d_out, int out_size, void* d_ws, size_t ws_size,
                              hipStream_t stream) {
  const float* pred   = (const float*)d_in[0];
  const int*   target = (const int*)d_in[1];
  float*       out    = (float*)d_out;

  const int B = in_sizes[1];            // 4096
  const int C = in_sizes[0] / B;        // 50257

  float* row_loss = (float*)d_ws;       // B floats of scratch

  if (C == 50257) {
    row_lse_kernel<50257><<<B, THREADS, 0, stream>>>(pred, target, row_loss, C);
  } else {
    row_lse_kernel<0><<<B, THREADS, 0, stream>>>(pred, target, row_loss, C);
  }
  mean_kernel<<<1, 256, 0, stream>>>(row_loss, out, B);
}